// DGCNN_773094114042
// MI455X (gfx1250) — compile-verified
//
#include <hip/hip_runtime.h>
#include <math.h>

#define NPTS 2048
#define BATCH 8
#define KNN 20
#define CATC 512
#define OUTC 1024

typedef float v2f __attribute__((ext_vector_type(2)));
typedef float v8f __attribute__((ext_vector_type(8)));

// ---------------------------------------------------------------------------
// FP32 WMMA 16x16x4: D = A(16x4)*B(4x16) + C, full fp32 (matches the float32
// reference). Layouts (ISA 7.12.2):
//   A 16x4 : lane l -> M=l&15, K={0,1}+2*(l>>4) in {a.x,a.y}
//   B 4x16 : lane l -> N=l&15, K={0,1}+2*(l>>4) in {b.x,b.y}
//   C/D    : vgpr v, lane l -> M = v + 8*(l>>4), N = l&15
// ---------------------------------------------------------------------------
__device__ __forceinline__ v8f wmma4(v2f a, v2f b, v8f c) {
#if __has_builtin(__builtin_amdgcn_wmma_f32_16x16x4_f32)
  return __builtin_amdgcn_wmma_f32_16x16x4_f32(false, a, false, b, (short)0, c,
                                               false, false);
#else
  const int l = threadIdx.x & 31;
  const int lm = l & 15;
  const float b0 = __shfl(b.x, lm, 32), b1 = __shfl(b.y, lm, 32);
  const float b2 = __shfl(b.x, lm + 16, 32), b3 = __shfl(b.y, lm + 16, 32);
#pragma unroll
  for (int v = 0; v < 8; ++v) {
    const int M = v + 8 * (l >> 4);
    const float a0 = __shfl(a.x, M, 32), a1 = __shfl(a.y, M, 32);
    const float a2 = __shfl(a.x, M + 16, 32), a3 = __shfl(a.y, M + 16, 32);
    c[v] += a0 * b0 + a1 * b1 + a2 * b2 + a3 * b3;
  }
  return c;
#endif
}

// ---------------------------------------------------------------------------
__global__ void sq_kernel(const float* __restrict__ feat, int stride, int C,
                          float* __restrict__ sq, int total) {
  const int i = blockIdx.x * blockDim.x + threadIdx.x;
  if (i < total) {
    const float* r = feat + (size_t)i * stride;
    float s = 0.f;
    for (int c = 0; c < C; ++c) s += r[c] * r[c];
    sq[i] = s;
  }
}

__global__ void zero_kernel(float* __restrict__ p, int n) {
  const int i = blockIdx.x * blockDim.x + threadIdx.x;
  if (i < n) p[i] = 0.f;
}

// ---------------------------------------------------------------------------
// d[n,m] = 2*<x_n,x_m> - |x_n|^2 - |x_m|^2 for batch b. One wave per 16x16
// tile; gram accumulated with f32 WMMA over compile-time C (unrolled, b64
// vector loads, no predication).
// ---------------------------------------------------------------------------
template <int C>
__global__ __launch_bounds__(32) void dist_kernel(
    const float* __restrict__ feat, int stride, const float* __restrict__ sq,
    float* __restrict__ dmat, int b) {
  const int l = threadIdx.x;
  const int half = l >> 4, lm = l & 15;
  const int colbase = blockIdx.x << 4;
  const int rowbase = blockIdx.y << 4;
  const float* fb = feat + (size_t)b * NPTS * stride;
  const float* sqb = sq + b * NPTS;
  const float* arow = fb + (size_t)(rowbase + lm) * stride;
  const float* brow = fb + (size_t)(colbase + lm) * stride;
  v8f acc = {0.f, 0.f, 0.f, 0.f, 0.f, 0.f, 0.f, 0.f};
  if constexpr (C == 3) {
    const float r0 = arow[0], r1 = arow[1], r2 = arow[2];
    const float s0 = brow[0], s1 = brow[1], s2 = brow[2];
    v2f a, bb;
    a.x = half ? r2 : r0;  a.y = half ? 0.f : r1;
    bb.x = half ? s2 : s0; bb.y = half ? 0.f : s1;
    acc = wmma4(a, bb, acc);
  } else {
#pragma unroll
    for (int c0 = 0; c0 < C; c0 += 4) {
      const int ca = c0 + 2 * half;
      const v2f a = *(const v2f*)(arow + ca);
      const v2f bb = *(const v2f*)(brow + ca);
      acc = wmma4(a, bb, acc);
    }
  }
  const float sqc = sqb[colbase + lm];
#pragma unroll
  for (int v = 0; v < 8; ++v) {
    const int M = v + 8 * half;
    dmat[(size_t)(rowbase + M) * NPTS + colbase + lm] =
        2.f * acc[v] - sqb[rowbase + M] - sqc;
  }
}

// ---------------------------------------------------------------------------
// Top-20 per row: 20 iterations of block-wide argmax over the LDS-resident
// row; ties toward lower index (matches jax.lax.top_k).
// ---------------------------------------------------------------------------
__global__ __launch_bounds__(256) void topk_kernel(
    const float* __restrict__ dmat, int* __restrict__ knn, int b) {
  __shared__ float row[NPTS];
  __shared__ float rv[256];
  __shared__ int ri[256];
  const int n = blockIdx.x, t = threadIdx.x;
  const float* dr = dmat + (size_t)n * NPTS;
  for (int i = t; i < NPTS; i += 256) row[i] = dr[i];
  __syncthreads();
  int* out = knn + ((size_t)b * NPTS + n) * KNN;
  for (int k = 0; k < KNN; ++k) {
    float bv = -__builtin_inff();
    int bi = NPTS;
    for (int i = t; i < NPTS; i += 256) {
      const float v = row[i];
      if (v > bv || (v == bv && i < bi)) { bv = v; bi = i; }
    }
    rv[t] = bv; ri[t] = bi;
    __syncthreads();
    for (int s = 128; s > 0; s >>= 1) {
      if (t < s) {
        const float ov = rv[t + s]; const int oi = ri[t + s];
        if (ov > rv[t] || (ov == rv[t] && oi < ri[t])) { rv[t] = ov; ri[t] = oi; }
      }
      __syncthreads();
    }
    if (t == 0) { out[k] = ri[0]; row[ri[0]] = -__builtin_inff(); }
    __syncthreads();
  }
}

// ---------------------------------------------------------------------------
// EdgeConv GEMM. One wave per (point p, 16-channel tile). Rows = k (20, padded
// to 2x16; padded rows carry garbage but are never consumed). K-dim = 2*CIN,
// split into an unconditional (feat-center) region and a (center) region.
// PASS 1: accumulate per-channel sum/sumsq of y over valid k.
// PASS 2: y -> BN -> leaky -> max over valid k, write to concat slice.
// ---------------------------------------------------------------------------
template <int PASS, int CIN>
__global__ __launch_bounds__(32) void edge_kernel(
    const float* __restrict__ feat, int stride, const int* __restrict__ knn,
    const float* __restrict__ w, int F, float* __restrict__ stats,
    const float* __restrict__ scale, const float* __restrict__ shift,
    float* __restrict__ outp, int ostride, int ooff) {
  __shared__ alignas(16) float center[CIN < 4 ? 4 : CIN];
  __shared__ int nb[KNN];
  const int p = blockIdx.x;      // global point index 0..B*N-1
  const int b = p >> 11;         // NPTS == 2048
  const int fbase = blockIdx.y << 4;
  const int l = threadIdx.x;
  const int half = l >> 4, lm = l & 15;
  constexpr int CP = 2 * CIN;

  const float* crow = feat + (size_t)p * stride;
  for (int c = l; c < CIN; c += 32) center[c] = crow[c];
  if (l < KNN) nb[l] = knn[(size_t)p * KNN + l];
  __syncthreads();

  const float* wrow = w + (size_t)(fbase + lm) * CP;

  float s1 = 0.f, s2 = 0.f;
  float mx = -__builtin_inff();
  float sc = 0.f, sh = 0.f;
  if (PASS == 2) { sc = scale[fbase + lm]; sh = shift[fbase + lm]; }

  for (int r = 0; r < 2; ++r) {
    const int k = 16 * r + lm;
    const int kc = (k < KNN) ? k : (KNN - 1);          // clamp for safe gather
    const float* np = feat + (size_t)((b << 11) + nb[kc]) * stride;
    __builtin_prefetch(np, 0, 1);
    v8f acc = {0.f, 0.f, 0.f, 0.f, 0.f, 0.f, 0.f, 0.f};
    if constexpr (CIN == 3) {
      const float h0 = np[0] - center[0];
      const float h1 = np[1] - center[1];
      const float h2 = np[2] - center[2];
      const float c0v = center[0], c1v = center[1], c2v = center[2];
      v2f a, bb;
      // c in {0,1} (half 0) / {2,3} (half 1)
      a.x = half ? h2 : h0;
      a.y = half ? c0v : h1;
      bb.x = wrow[2 * half];
      bb.y = wrow[2 * half + 1];
      acc = wmma4(a, bb, acc);
      // c in {4,5} (half 0) / {6,7}=pad (half 1)
      a.x = half ? 0.f : c1v;
      a.y = half ? 0.f : c2v;
      bb.x = half ? 0.f : wrow[4];
      bb.y = half ? 0.f : wrow[5];
      acc = wmma4(a, bb, acc);
    } else {
#pragma unroll
      for (int c0 = 0; c0 < CIN; c0 += 4) {            // region 1: feat-center
        const int ca = c0 + 2 * half;
        const v2f xn = *(const v2f*)(np + ca);
        const v2f xc = *(const v2f*)(&center[ca]);
        const v2f a = xn - xc;
        const v2f bb = *(const v2f*)(wrow + ca);
        acc = wmma4(a, bb, acc);
      }
#pragma unroll
      for (int c0 = 0; c0 < CIN; c0 += 4) {            // region 2: center
        const int ca = c0 + 2 * half;
        const v2f a = *(const v2f*)(&center[ca]);
        const v2f bb = *(const v2f*)(wrow + CIN + ca);
        acc = wmma4(a, bb, acc);
      }
    }
#pragma unroll
    for (int v = 0; v < 8; ++v) {
      const int kk = 16 * r + v + 8 * half;
      if (kk < KNN) {
        const float yv = acc[v];
        if (PASS == 1) {
          s1 += yv; s2 += yv * yv;
        } else {
          float t = sc * yv + sh;
          t = (t >= 0.f) ? t : 0.2f * t;
          mx = fmaxf(mx, t);
        }
      }
    }
  }
  if (PASS == 1) {
    s1 += __shfl_xor(s1, 16, 32);
    s2 += __shfl_xor(s2, 16, 32);
    if (l < 16) {
      atomicAdd(&stats[fbase + lm], s1);
      atomicAdd(&stats[F + fbase + lm], s2);
    }
  } else {
    mx = fmaxf(mx, __shfl_xor(mx, 16, 32));
    if (l < 16) outp[(size_t)p * ostride + ooff + fbase + lm] = mx;
  }
}

// ---------------------------------------------------------------------------
__global__ void finalize_kernel(const float* __restrict__ stats,
                                const float* __restrict__ g,
                                const float* __restrict__ bb,
                                float* __restrict__ scale,
                                float* __restrict__ shift, int F, float count) {
  const int f = blockIdx.x * blockDim.x + threadIdx.x;
  if (f < F) {
    const float mean = stats[f] / count;
    const float var = stats[F + f] / count - mean * mean;
    const float sc = g[f] * rsqrtf(var + 1e-5f);
    scale[f] = sc;
    shift[f] = bb[f] - mean * sc;
  }
}

// ---------------------------------------------------------------------------
// Final 512->1024 pointwise conv: 16 points x 16 channels per wave via WMMA.
// ---------------------------------------------------------------------------
__global__ __launch_bounds__(32) void fconv_kernel(
    const float* __restrict__ catf, const float* __restrict__ wf,
    float* __restrict__ y, float* __restrict__ stats) {
  const int pbase = blockIdx.x << 4, fbase = blockIdx.y << 4;
  const int l = threadIdx.x, half = l >> 4, lm = l & 15;
  const float* ar = catf + (size_t)(pbase + lm) * CATC;
  const float* wr = wf + (size_t)(fbase + lm) * CATC;
  v8f acc = {0.f, 0.f, 0.f, 0.f, 0.f, 0.f, 0.f, 0.f};
#pragma unroll 8
  for (int c0 = 0; c0 < CATC; c0 += 4) {
    const int ca = c0 + 2 * half;
    const v2f a = *(const v2f*)(ar + ca);
    const v2f bb = *(const v2f*)(wr + ca);
    acc = wmma4(a, bb, acc);
  }
  float s1 = 0.f, s2 = 0.f;
#pragma unroll
  for (int v = 0; v < 8; ++v) {
    const int p = pbase + v + 8 * half;
    const float val = acc[v];
    y[(size_t)p * OUTC + fbase + lm] = val;
    s1 += val; s2 += val * val;
  }
  s1 += __shfl_xor(s1, 16, 32);
  s2 += __shfl_xor(s2, 16, 32);
  if (l < 16) {
    atomicAdd(&stats[fbase + lm], s1);
    atomicAdd(&stats[OUTC + fbase + lm], s2);
  }
}

__global__ void fapply_kernel(float* __restrict__ y,
                              const float* __restrict__ scale,
                              const float* __restrict__ shift, size_t total) {
  const size_t i = (size_t)blockIdx.x * blockDim.x + threadIdx.x;
  if (i < total) {
    const int f = (int)(i & (OUTC - 1));
    float v = scale[f] * y[i] + shift[f];
    y[i] = (v >= 0.f) ? v : 0.2f * v;
  }
}

// ---------------------------------------------------------------------------
template <int CIN>
static void run_layer(const float* featIn, int stride, const float* w,
                      const float* g, const float* bb, int F, float* sq,
                      float* dmat, int* knn, float* stats, float* scale,
                      float* shift, float* catf, int ooff, hipStream_t stream) {
  const int P = BATCH * NPTS;
  sq_kernel<<<(P + 255) / 256, 256, 0, stream>>>(featIn, stride, CIN, sq, P);
  for (int b = 0; b < BATCH; ++b) {
    dist_kernel<CIN><<<dim3(NPTS / 16, NPTS / 16), 32, 0, stream>>>(
        featIn, stride, sq, dmat, b);
    topk_kernel<<<NPTS, 256, 0, stream>>>(dmat, knn, b);
  }
  zero_kernel<<<(2 * F + 255) / 256, 256, 0, stream>>>(stats, 2 * F);
  dim3 eg(P, F / 16);
  edge_kernel<1, CIN><<<eg, 32, 0, stream>>>(featIn, stride, knn, w, F, stats,
                                             nullptr, nullptr, nullptr, 0, 0);
  finalize_kernel<<<(F + 255) / 256, 256, 0, stream>>>(
      stats, g, bb, scale, shift, F, (float)((size_t)P * KNN));
  edge_kernel<2, CIN><<<eg, 32, 0, stream>>>(featIn, stride, knn, w, F, nullptr,
                                             scale, shift, catf, CATC, ooff);
}

extern "C" void kernel_launch(void* const* d_in, const int* in_sizes, int n_in,
                              void* d_out, int out_size, void* d_ws,
                              size_t ws_size, hipStream_t stream) {
  (void)in_sizes; (void)n_in; (void)out_size; (void)ws_size;
  const float* x = (const float*)d_in[0];
  const float* W[4]; const float* G[4]; const float* Bb[4];
  for (int i = 0; i < 4; ++i) {
    W[i]  = (const float*)d_in[1 + 3 * i];
    G[i]  = (const float*)d_in[2 + 3 * i];
    Bb[i] = (const float*)d_in[3 + 3 * i];
  }
  const float* wf = (const float*)d_in[13];
  const float* gf = (const float*)d_in[14];
  const float* bf = (const float*)d_in[15];

  char* wsb = (char*)d_ws;
  float* dmat  = (float*)wsb; wsb += (size_t)NPTS * NPTS * 4;          // 16 MB
  float* sq    = (float*)wsb; wsb += (size_t)BATCH * NPTS * 4;
  int*   knn   = (int*)wsb;   wsb += (size_t)BATCH * NPTS * KNN * 4;
  float* catf  = (float*)wsb; wsb += (size_t)BATCH * NPTS * CATC * 4;  // 32 MB
  float* stats = (float*)wsb; wsb += (size_t)2 * OUTC * 4;
  float* scale = (float*)wsb; wsb += (size_t)OUTC * 4;
  float* shift = (float*)wsb; wsb += (size_t)OUTC * 4;

  const int P = BATCH * NPTS;

  run_layer<3>(x, 3, W[0], G[0], Bb[0], 64, sq, dmat, knn, stats, scale, shift,
               catf, 0, stream);
  run_layer<64>(catf + 0, CATC, W[1], G[1], Bb[1], 64, sq, dmat, knn, stats,
                scale, shift, catf, 64, stream);
  run_layer<64>(catf + 64, CATC, W[2], G[2], Bb[2], 128, sq, dmat, knn, stats,
                scale, shift, catf, 128, stream);
  run_layer<128>(catf + 128, CATC, W[3], G[3], Bb[3], 256, sq, dmat, knn, stats,
                 scale, shift, catf, 256, stream);

  float* y = (float*)d_out;
  zero_kernel<<<(2 * OUTC + 255) / 256, 256, 0, stream>>>(stats, 2 * OUTC);
  fconv_kernel<<<dim3(P / 16, OUTC / 16), 32, 0, stream>>>(catf, wf, y, stats);
  finalize_kernel<<<(OUTC + 255) / 256, 256, 0, stream>>>(
      stats, gf, bf, scale, shift, OUTC, (float)P);
  const size_t tot = (size_t)P * OUTC;
  fapply_kernel<<<(tot + 255) / 256, 256, 0, stream>>>(y, scale, shift, tot);
}